// MomaGraphTokenizer_29609504539321
// MI455X (gfx1250) — compile-verified
//
#include <hip/hip_runtime.h>
#include <hip/hip_bf16.h>
#include <math.h>

// ---------------------------------------------------------------------------
// MomaGraphTokenizer fused kernel for gfx1250 (MI455X).
// Memory-bound (~130MB HBM floor, inputs L2-resident). fp32 WMMA 16x16x4 for
// all three linear layers. Two waves per block cooperate on a PAIR of
// 16-token tiles and split the N-tiles, so every packed-B fragment load from
// L2 feeds two v_wmma ops (halves the dominant L2 traffic vs 1 tile/wave).
// ---------------------------------------------------------------------------

typedef float v2f __attribute__((ext_vector_type(2)));
typedef float v8f __attribute__((ext_vector_type(8)));

#define Bc         8
#define NCc        8
#define Lc         1024
#define NFEATc     256
#define MAX_OBJSc  512
#define MAX_EDGESc 4096
#define NIDc       128
#define OUTDIMc    512

#define WMMA_F32(a, b, c) \
  __builtin_amdgcn_wmma_f32_16x16x4_f32(false, (a), false, (b), (short)0, (c), false, false)

// Workspace layout (floats):
//   [0..63]  per-clip max of tok_t_idxs
//   then     attrP packed: 16 nt * 64 kc * 32 lanes * float2 = 32768 float2
//   then     bboxP packed:  4 nt *  4 kc * 32 lanes * float2 =   512 float2
//   then     nidP  packed:  4 nt * 64 kc * 32 lanes * float2 =  8192 float2
#define WS_ATTRP_F2 32768
#define WS_BBOXP_F2 512
#define WS_NIDP_F2  8192

// ---------------------------------------------------------------------------
// Kernel 1: per-clip max over 2048 time values
// ---------------------------------------------------------------------------
__global__ void moma_clipmax_kernel(const float* __restrict__ tok_t, float* __restrict__ ws) {
  __shared__ float red[256];
  const int clip = blockIdx.x;
  const float* p = tok_t + (size_t)clip * (2 * Lc);
  float m = -INFINITY;
  for (int i = threadIdx.x; i < 2 * Lc; i += 256) m = fmaxf(m, p[i]);
  red[threadIdx.x] = m;
  __syncthreads();
  for (int s = 128; s > 0; s >>= 1) {
    if ((int)threadIdx.x < s) red[threadIdx.x] = fmaxf(red[threadIdx.x], red[threadIdx.x + s]);
    __syncthreads();
  }
  if (threadIdx.x == 0) ws[clip] = red[0];
}

// ---------------------------------------------------------------------------
// Kernel 2: pack weights into WMMA B-fragment order.
// B (4x16, KxN): VGPR0 lanes0-15 = K=k0, lanes16-31 = K=k0+2; VGPR1 = +1/+3.
// Packed index = ((nt*chunks + chunk)*32 + lane) -> float2{W[k][n], W[k+1][n]}
// ---------------------------------------------------------------------------
__device__ __forceinline__ void pack_one(const float* __restrict__ W, float2* __restrict__ P,
                                         int chunks, int N, int f) {
  const int lane  = f & 31;
  const int fi    = f >> 5;
  const int chunk = fi % chunks;
  const int nt    = fi / chunks;
  const int col   = nt * 16 + (lane & 15);
  const int krow  = chunk * 4 + ((lane >> 4) << 1);
  P[f] = make_float2(W[krow * N + col], W[(krow + 1) * N + col]);
}

__global__ void moma_pack_kernel(const float* __restrict__ aW, const float* __restrict__ bW,
                                 const float* __restrict__ nW, float* __restrict__ ws) {
  const int tid = blockIdx.x * blockDim.x + threadIdx.x;
  float2* attrP = (float2*)(ws + 64);
  float2* bboxP = attrP + WS_ATTRP_F2;
  float2* nidP  = bboxP + WS_BBOXP_F2;
  if (tid < WS_ATTRP_F2) {
    pack_one(aW, attrP, 64, 256, tid);                           // attr_W [256x256]
  } else if (tid < WS_ATTRP_F2 + WS_BBOXP_F2) {
    pack_one(bW, bboxP, 4, 64, tid - WS_ATTRP_F2);               // bbox_W [16x64]
  } else if (tid < WS_ATTRP_F2 + WS_BBOXP_F2 + WS_NIDP_F2) {
    pack_one(nW, nidP, 64, 64, tid - WS_ATTRP_F2 - WS_BBOXP_F2); // n_id_W [256x64]
  }
}

// ---------------------------------------------------------------------------
// Kernel 3: fused tokenizer. 64 threads = 2 waves per block; the block owns
// TWO 16-token tiles. A-tiles live in LDS in WMMA *fragment order*
// ([kc][lane]{k,k+1}) -> ds_load_b64 is bank-conflict-free with no padding.
// Each wave computes half the N-tiles for BOTH tiles: 1 B load : 2 WMMAs.
// ---------------------------------------------------------------------------
__global__ __launch_bounds__(64) void moma_main_kernel(
    const int* __restrict__ tok_n_idxs, const float* __restrict__ tok_t,
    const int* __restrict__ tok_types,  const int* __restrict__ tok_e,
    const float* __restrict__ nfeats,   const float* __restrict__ efeats,
    const float* __restrict__ bbox_feats, const int* __restrict__ idx_lup,
    const float* __restrict__ nid_lup,
    const float* __restrict__ attr_b,   const float* __restrict__ bbox_b,
    const float* __restrict__ freq,     const float* __restrict__ phs,
    const float* __restrict__ nid_b,    const float* __restrict__ type_emb,
    const float* __restrict__ ws,       float* __restrict__ out)
{
  __shared__ float smA[2 * 16 * NFEATc];   // 2 tiles * 4096 dwords = 32 KB

  const int wave    = threadIdx.x >> 5;
  const int lane    = threadIdx.x & 31;
  const int token00 = blockIdx.x * 32;     // first token of the tile pair
  const int clip    = token00 >> 10;       // / L
  const int b       = clip >> 3;           // / NC
  const int l00     = token00 & (Lc - 1);

  const float  clipmax = ws[clip];
  const float2* attrP = (const float2*)(ws + 64);
  const float2* bboxP = attrP + WS_ATTRP_F2;
  const float2* nidP  = bboxP + WS_BBOXP_F2;

  // gather roles: wave = tile-in-pair, lane -> (row, half)
  const int gt = wave, gr = lane >> 1, gh = lane & 1;

  // ---- gather attr features (nfeats for types 0/2, efeats for 1) into LDS ----
  {
    const int tok = token00 + gt * 16 + gr;
    const int typ = tok_types[tok];
    const bool nonedge = (typ == 0) | (typ == 2);
    const int nidx = tok_n_idxs[2 * tok];
    const int eidx = tok_e[tok];
    const float* src = nonedge
        ? nfeats + ((size_t)b * MAX_OBJSc  + nidx) * NFEATc
        : efeats + ((size_t)b * MAX_EDGESc + eidx) * NFEATc;
    const float4* g4 = (const float4*)(src + gh * 128);
    float* base = smA + gt * 4096 + gr * 2;
#pragma unroll 8
    for (int i = 0; i < 32; ++i) {       // store fragment-order: [kc][lane]{k,k+1}
      float4 v = g4[i];
      float* d = base + (gh * 32 + i) * 64;
      *(float2*)(d)      = make_float2(v.x, v.y);   // rows -> lanes 0..15
      *(float2*)(d + 32) = make_float2(v.z, v.w);   // k+2,k+3 -> lanes 16..31
    }
  }
  __syncthreads();

  const int arow  = lane & 15;
  const int koff  = (lane >> 4) << 1;
  const int mbase = (lane >> 4) << 3;      // C/D rows: lanes16-31 hold rows 8..15

  int ty[2][8];                            // token types for epilogue (L2-resident)
#pragma unroll
  for (int t = 0; t < 2; ++t)
#pragma unroll
    for (int j = 0; j < 8; ++j) ty[t][j] = tok_types[token00 + t * 16 + mbase + j];

  float* outrow = out + (size_t)token00 * OUTDIMc;

  // ---- attr: [16x256] @ [256x256] -> out[:, 0:256]; wave handles 8 N-tiles ----
#pragma unroll 1
  for (int ni = 0; ni < 8; ++ni) {
    const int nt = wave * 8 + ni;
    v8f acc0 = {}, acc1 = {};
    const float2* bp = attrP + nt * 64 * 32 + lane;
    const float* a0 = smA + lane * 2;
    const float* a1 = smA + 4096 + lane * 2;
#pragma unroll 8
    for (int kc = 0; kc < 64; ++kc) {
      v2f bf = *(const v2f*)(bp + kc * 32);
      v2f x0 = *(const v2f*)(a0 + kc * 64);
      v2f x1 = *(const v2f*)(a1 + kc * 64);
      acc0 = WMMA_F32(x0, bf, acc0);
      acc1 = WMMA_F32(x1, bf, acc1);
    }
    const int col = nt * 16 + arow;
    const float bias = attr_b[col];
#pragma unroll
    for (int j = 0; j < 8; ++j) {
      outrow[(size_t)(mbase + j) * OUTDIMc + col] =
          acc0[j] + bias + type_emb[ty[0][j] * OUTDIMc + col];
      outrow[(size_t)(16 + mbase + j) * OUTDIMc + col] =
          acc1[j] + bias + type_emb[ty[1][j] * OUTDIMc + col];
    }
  }

  // ---- bbox: [16x16] @ [16x64] -> out[:, 256:320]; wave handles 2 N-tiles ----
  {
    const float* bb = bbox_feats + (size_t)token00 * 16;
#pragma unroll 1
    for (int ni = 0; ni < 2; ++ni) {
      const int nt = wave * 2 + ni;
      v8f acc0 = {}, acc1 = {};
      const float2* bp = bboxP + nt * 4 * 32 + lane;
#pragma unroll
      for (int kc = 0; kc < 4; ++kc) {
        v2f bf = *(const v2f*)(bp + kc * 32);
        v2f x0 = *(const v2f*)&bb[arow * 16 + kc * 4 + koff];
        v2f x1 = *(const v2f*)&bb[256 + arow * 16 + kc * 4 + koff];
        acc0 = WMMA_F32(x0, bf, acc0);
        acc1 = WMMA_F32(x1, bf, acc1);
      }
      const int col = nt * 16 + arow;
      const float bias = bbox_b[col];
#pragma unroll
      for (int j = 0; j < 8; ++j) {
        outrow[(size_t)(mbase + j) * OUTDIMc + 256 + col] =
            acc0[j] + bias + type_emb[ty[0][j] * OUTDIMc + 256 + col];
        outrow[(size_t)(16 + mbase + j) * OUTDIMc + 256 + col] =
            acc1[j] + bias + type_emb[ty[1][j] * OUTDIMc + 256 + col];
      }
    }
  }

  // ---- time encoding: cos((clipmax - t_j)*freq + phase) -> out[:, 320:448] ----
  {
    const int r2 = lane >> 1, j2 = lane & 1;
    const int tok = token00 + wave * 16 + r2;   // each wave covers its own tile
    const float ph = clipmax - tok_t[2 * tok + j2];
    const int typ = tok_types[tok];
    const float* emb = type_emb + typ * OUTDIMc + 320 + j2 * 64;
    float* orow = out + (size_t)tok * OUTDIMc + 320 + j2 * 64;
#pragma unroll 4
    for (int e = 0; e < 64; ++e)
      orow[e] = __cosf(ph * freq[e] + phs[e]) + emb[e];
  }

  __syncthreads();   // all attr LDS reads done before nid gather overwrites

  // ---- nid gather: two 128-wide n_id_lookup rows per token, fragment order ----
  {
    const int lpos = l00 + gt * 16 + gr;
    const int idx = idx_lup[(size_t)clip * (2 * Lc) + 2 * lpos + gh];
    const float4* g4 = (const float4*)(nid_lup + ((size_t)clip * NIDc + idx) * NIDc);
    float* base = smA + gt * 4096 + gr * 2;
#pragma unroll 8
    for (int i = 0; i < 32; ++i) {
      float4 v = g4[i];
      float* d = base + (gh * 32 + i) * 64;
      *(float2*)(d)      = make_float2(v.x, v.y);
      *(float2*)(d + 32) = make_float2(v.z, v.w);
    }
  }
  __syncthreads();

  // ---- nid: [16x256] @ [256x64] -> out[:, 448:512]; wave handles 2 N-tiles ----
#pragma unroll 1
  for (int ni = 0; ni < 2; ++ni) {
    const int nt = wave * 2 + ni;
    v8f acc0 = {}, acc1 = {};
    const float2* bp = nidP + nt * 64 * 32 + lane;
    const float* a0 = smA + lane * 2;
    const float* a1 = smA + 4096 + lane * 2;
#pragma unroll 8
    for (int kc = 0; kc < 64; ++kc) {
      v2f bf = *(const v2f*)(bp + kc * 32);
      v2f x0 = *(const v2f*)(a0 + kc * 64);
      v2f x1 = *(const v2f*)(a1 + kc * 64);
      acc0 = WMMA_F32(x0, bf, acc0);
      acc1 = WMMA_F32(x1, bf, acc1);
    }
    const int col = nt * 16 + arow;
    const float bias = nid_b[col];
#pragma unroll
    for (int j = 0; j < 8; ++j) {
      outrow[(size_t)(mbase + j) * OUTDIMc + 448 + col] =
          acc0[j] + bias + type_emb[ty[0][j] * OUTDIMc + 448 + col];
      outrow[(size_t)(16 + mbase + j) * OUTDIMc + 448 + col] =
          acc1[j] + bias + type_emb[ty[1][j] * OUTDIMc + 448 + col];
    }
  }
}

// ---------------------------------------------------------------------------
extern "C" void kernel_launch(void* const* d_in, const int* in_sizes, int n_in,
                              void* d_out, int out_size, void* d_ws, size_t ws_size,
                              hipStream_t stream) {
  (void)in_sizes; (void)n_in; (void)out_size; (void)ws_size;
  const int*   tok_n  = (const int*)  d_in[0];
  const float* tok_t  = (const float*)d_in[1];
  const int*   tok_ty = (const int*)  d_in[2];
  const int*   tok_e  = (const int*)  d_in[3];
  const float* nfeats = (const float*)d_in[4];
  const float* efeats = (const float*)d_in[5];
  const float* bbox   = (const float*)d_in[6];
  const int*   idxl   = (const int*)  d_in[7];
  const float* nidl   = (const float*)d_in[8];
  const float* attrW  = (const float*)d_in[9];
  const float* attrB  = (const float*)d_in[10];
  const float* bboxW  = (const float*)d_in[11];
  const float* bboxB  = (const float*)d_in[12];
  const float* freq   = (const float*)d_in[13];
  const float* phs    = (const float*)d_in[14];
  const float* nidW   = (const float*)d_in[15];
  const float* nidB   = (const float*)d_in[16];
  const float* temb   = (const float*)d_in[17];
  float* out = (float*)d_out;
  float* ws  = (float*)d_ws;

  moma_clipmax_kernel<<<Bc * NCc, 256, 0, stream>>>(tok_t, ws);
  moma_pack_kernel<<<(WS_ATTRP_F2 + WS_BBOXP_F2 + WS_NIDP_F2 + 255) / 256, 256, 0, stream>>>(
      attrW, bboxW, nidW, ws);

  const int tiles = Bc * NCc * Lc / 16;    // 4096 tiles, 2 tiles per block
  moma_main_kernel<<<tiles / 2, 64, 0, stream>>>(
      tok_n, tok_t, tok_ty, tok_e, nfeats, efeats, bbox, idxl, nidl,
      attrB, bboxB, freq, phs, nidB, temb, ws, out);
}